// FISTA_7550552506950
// MI455X (gfx1250) — compile-verified
//
#include <hip/hip_runtime.h>
#include <math.h>

#define SIGNAL_DIM 64
#define FR_SIZE    4096
#define BATCH      1024
#define MAX_ITER   25
#define REG_T      0.5f

typedef __attribute__((ext_vector_type(2))) float v2f;
typedef __attribute__((ext_vector_type(4))) float v4f;
typedef __attribute__((ext_vector_type(8))) float v8f;

// D = A(16x4, f32) x B(4x16, f32) + C(16x16, f32)
static __device__ __forceinline__ v8f wmma4(v2f a, v2f b, v8f c) {
  // (neg_a, A, neg_b, B, c_mod, C, reuse_a, reuse_b); NEG[1:0] must be 0 for f32
  return __builtin_amdgcn_wmma_f32_16x16x4_f32(false, a, false, b, (short)0, c, false, false);
}

// De-interleave input dictionary into planar D_re/D_im ([64][4096]) and
// planar transposed Dt_re/Dt_im ([4096][64]).
__global__ void fista_prep(const float* __restrict__ Din,
                           float* __restrict__ D_re, float* __restrict__ D_im,
                           float* __restrict__ Dt_re, float* __restrict__ Dt_im) {
  int idx = blockIdx.x * blockDim.x + threadIdx.x;   // over 64*4096 complex elems
  if (idx < SIGNAL_DIM * FR_SIZE) {
    int n = idx / FR_SIZE, f = idx % FR_SIZE;
    float re = Din[2 * idx], im = Din[2 * idx + 1];
    D_re[idx] = re;
    D_im[idx] = im;
    Dt_re[f * SIGNAL_DIM + n] = re;
    Dt_im[f * SIGNAL_DIM + n] = im;
  }
}

// One workgroup = 16 batch rows, fully fused 25 FISTA iterations.
__global__ void __launch_bounds__(256) fista_kernel(
    const float* __restrict__ X,      // [1024][2][64] f32
    const float* __restrict__ D_re,   // [64][4096]
    const float* __restrict__ D_im,
    const float* __restrict__ Dt_re,  // [4096][64]
    const float* __restrict__ Dt_im,
    float* __restrict__ W_re,         // [1024][4096] planar workspaces
    float* __restrict__ W_im,
    float* __restrict__ Z_re,
    float* __restrict__ Z_im,
    float* __restrict__ out,          // [1024][4096] magnitude (unnormalized)
    float* __restrict__ wsmax)        // [64] per-block max
{
  __shared__ float s_part[4][16][16][2];  // khalf=1 partial resid tiles
  __shared__ float s_res_re[16][64];      // final residual, [m][n]
  __shared__ float s_res_im[16][64];
  __shared__ float s_red[256];

  const int tid  = threadIdx.x;
  const int lane = tid & 31;
  const int wid  = tid >> 5;     // 0..7
  const int j    = lane & 15;    // N/M position within half-wave
  const int h    = lane >> 4;    // half-wave select (K pair / M+8)
  const int b0   = blockIdx.x * 16;
  const size_t rowbase = (size_t)b0 * FR_SIZE;  // block's rows are contiguous

  // zero-init this block's rows of W and Z (harness poisons ws)
  {
    v4f zz = {0.0f, 0.0f, 0.0f, 0.0f};
    for (int i = tid; i < 16 * FR_SIZE / 4; i += 256) {
      *(v4f*)&W_re[rowbase + 4 * i] = zz;
      *(v4f*)&W_im[rowbase + 4 * i] = zz;
      *(v4f*)&Z_re[rowbase + 4 * i] = zz;
      *(v4f*)&Z_im[rowbase + 4 * i] = zz;
    }
  }
  __threadfence();
  __syncthreads();

  const float step = 1.0f / (float)FR_SIZE;
  const float thr  = REG_T * step;
  float t = 1.0f;

  const int ntile = wid & 3;   // which 16-wide N tile of resid (N=64 -> 4 tiles)
  const int khalf = wid >> 2;  // K=4096 split in halves across wave pairs
  const int n0    = ntile * 16;

  float a2re[32], a2im[32];    // phase-2 A operand (resid) held in registers

  for (int it = 0; it < MAX_ITER; ++it) {
    float t_new = 0.5f * (1.0f + sqrtf(1.0f + 4.0f * t * t));
    float mom   = (t - 1.0f) / t_new;
    t = t_new;

    // ---------- Phase 1: resid[16,64] = w @ D^T - xc ----------
    v8f rr = {}, ii = {}, ri = {}, ir = {};
    const float* wrre = &W_re[(size_t)(b0 + j) * FR_SIZE];
    const float* wrim = &W_im[(size_t)(b0 + j) * FR_SIZE];
    const float* drre = &D_re[(size_t)(n0 + j) * FR_SIZE];
    const float* drim = &D_im[(size_t)(n0 + j) * FR_SIZE];
    const int kbeg = khalf * (FR_SIZE / 2);
    for (int k = kbeg; k < kbeg + FR_SIZE / 2; k += 4) {
      int o = k + 2 * h;
      v2f are = *(const v2f*)&wrre[o];   // direct b64 -> even VGPR pair, no shuffles
      v2f aim = *(const v2f*)&wrim[o];
      v2f bre = *(const v2f*)&drre[o];
      v2f bim = *(const v2f*)&drim[o];
      rr = wmma4(are, bre, rr);
      ii = wmma4(aim, bim, ii);
      ri = wmma4(are, bim, ri);
      ir = wmma4(aim, bre, ir);
    }
    v8f res_re = rr - ii;   // (a+bi)(c+di): re = ac - bd
    v8f res_im = ri + ir;   //               im = ad + bc

    if (khalf == 1) {
#pragma unroll
      for (int v = 0; v < 8; ++v) {
        int m = v + 8 * h;
        s_part[ntile][m][j][0] = res_re[v];
        s_part[ntile][m][j][1] = res_im[v];
      }
    }
    __syncthreads();
    if (khalf == 0) {
#pragma unroll
      for (int v = 0; v < 8; ++v) {
        int m = v + 8 * h;
        int b = b0 + m;
        int n = n0 + j;
        float re = res_re[v] + s_part[ntile][m][j][0] - X[(size_t)b * 128 + n];
        float im = res_im[v] + s_part[ntile][m][j][1] - X[(size_t)b * 128 + 64 + n];
        s_res_re[m][n] = re;
        s_res_im[m][n] = im;
      }
    }
    __syncthreads();

    // Load phase-2 A operand (A[m][k], m=j, k over 64) into registers
#pragma unroll
    for (int kk = 0; kk < 16; ++kk) {
      int kb = 4 * kk + 2 * h;
      a2re[2 * kk]     = s_res_re[j][kb];
      a2re[2 * kk + 1] = s_res_re[j][kb + 1];
      a2im[2 * kk]     = s_res_im[j][kb];
      a2im[2 * kk + 1] = s_res_im[j][kb + 1];
    }

    // ---------- Phase 2: g = resid @ conj(D); soft-threshold + momentum ----------
    for (int tile = wid; tile < FR_SIZE / 16; tile += 8) {
      int f0 = tile * 16;
      v8f prr = {}, pii = {}, pri = {}, pir = {};
      const float* dtre = &Dt_re[(size_t)(f0 + j) * SIGNAL_DIM];
      const float* dtim = &Dt_im[(size_t)(f0 + j) * SIGNAL_DIM];
#pragma unroll
      for (int kk = 0; kk < 16; ++kk) {
        int o = 4 * kk + 2 * h;
        v2f bre = *(const v2f*)&dtre[o];
        v2f bim = *(const v2f*)&dtim[o];
        v2f are = {a2re[2 * kk], a2re[2 * kk + 1]};
        v2f aim = {a2im[2 * kk], a2im[2 * kk + 1]};
        prr = wmma4(are, bre, prr);
        pii = wmma4(aim, bim, pii);
        pri = wmma4(are, bim, pri);
        pir = wmma4(aim, bre, pir);
      }
      // conj(D): g_re = sum a_re*D_re + a_im*D_im ; g_im = sum a_im*D_re - a_re*D_im
      v8f g_re = prr + pii;
      v8f g_im = pir - pri;
#pragma unroll
      for (int v = 0; v < 8; ++v) {
        int b = b0 + v + 8 * h;
        int f = f0 + j;
        size_t idx = (size_t)b * FR_SIZE + f;
        float ur = W_re[idx] - step * g_re[v];
        float ui = W_im[idx] - step * g_im[v];
        float mag = sqrtf(ur * ur + ui * ui);
        float scale = (mag > thr) ? (mag - thr) / fmaxf(mag, 1e-12f) : 0.0f;
        float znr = ur * scale, zni = ui * scale;
        float zor = Z_re[idx], zoi = Z_im[idx];
        float wnr = znr + mom * (znr - zor);
        float wni = zni + mom * (zni - zoi);
        Z_re[idx] = znr;
        Z_im[idx] = zni;
        W_re[idx] = wnr;
        W_im[idx] = wni;
      }
    }
    __threadfence();   // make W/Z stores visible past WGP$ for next iteration
    __syncthreads();
  }

  // ---------- Finalize: mag = |z|, per-block max ----------
  float tmax = 0.0f;
  for (int i = tid; i < 16 * FR_SIZE; i += 256) {
    size_t idx = rowbase + i;
    float zr = Z_re[idx], zi = Z_im[idx];
    float m = sqrtf(zr * zr + zi * zi);
    out[idx] = m;
    tmax = fmaxf(tmax, m);
  }
  s_red[tid] = tmax;
  __syncthreads();
  for (int s = 128; s > 0; s >>= 1) {
    if (tid < s) s_red[tid] = fmaxf(s_red[tid], s_red[tid + s]);
    __syncthreads();
  }
  if (tid == 0) wsmax[blockIdx.x] = s_red[0];
}

__global__ void fista_norm(float* __restrict__ out, const float* __restrict__ wsmax) {
  __shared__ float s_gmax;
  if (threadIdx.x == 0) {
    float m = 0.0f;
    for (int i = 0; i < BATCH / 16; ++i) m = fmaxf(m, wsmax[i]);
    s_gmax = m;
  }
  __syncthreads();
  float g = s_gmax;
  int i = blockIdx.x * blockDim.x + threadIdx.x;
  if (i < BATCH * FR_SIZE) out[i] = out[i] / g;
}

extern "C" void kernel_launch(void* const* d_in, const int* in_sizes, int n_in,
                              void* d_out, int out_size, void* d_ws, size_t ws_size,
                              hipStream_t stream) {
  const float* X  = (const float*)d_in[0];   // [1024,2,64] f32
  const float* Dg = (const float*)d_in[1];   // [64,4096] complex64 interleaved

  const size_t DN = (size_t)SIGNAL_DIM * FR_SIZE;  // 262144
  const size_t BF = (size_t)BATCH * FR_SIZE;       // 4194304

  float* ws    = (float*)d_ws;
  float* D_re  = ws;
  float* D_im  = D_re + DN;
  float* Dt_re = D_im + DN;
  float* Dt_im = Dt_re + DN;
  float* W_re  = Dt_im + DN;
  float* W_im  = W_re + BF;
  float* Z_re  = W_im + BF;
  float* Z_im  = Z_re + BF;
  float* wsmax = Z_im + BF;   // 64 floats
  float* out   = (float*)d_out;

  fista_prep<<<(int)((DN + 255) / 256), 256, 0, stream>>>(Dg, D_re, D_im, Dt_re, Dt_im);
  fista_kernel<<<BATCH / 16, 256, 0, stream>>>(X, D_re, D_im, Dt_re, Dt_im,
                                               W_re, W_im, Z_re, Z_im, out, wsmax);
  fista_norm<<<(int)((BF + 255) / 256), 256, 0, stream>>>(out, wsmax);
}